// Rasterizer_1108101563060
// MI455X (gfx1250) — compile-verified
//
#include <hip/hip_runtime.h>
#include <stdint.h>

// Problem constants (match reference)
#define Hh 512
#define Ww 512
#define Gg 83
#define Kk 12
#define Nn 8
#define Vv (Gg * Gg)                 // 6889 vertices
#define Ff (2 * (Gg - 1) * (Gg - 1)) // 13448 faces
#define HW (Hh * Ww)                 // 262144 pixels (2^18)
#define FB 8                         // faces per block
#define FBLK (Ff / FB)               // 1681 blocks per image (exact)

typedef unsigned long long u64;
typedef unsigned int u32;

// ---- gfx1250 async global->LDS path (guarded; falls back to plain loads) ----
#if defined(__has_builtin)
#if __has_builtin(__builtin_amdgcn_global_load_async_to_lds_b32)
#define HAVE_ASYNC_LDS 1
#endif
#if __has_builtin(__builtin_amdgcn_s_wait_asynccnt)
#define HAVE_WAIT_ASYNC 1
#endif
#endif

// ---------------------------------------------------------------------------
// Kernel 1: re-initialize the z-buffer (u64 sentinel) every launch (graph-safe)
// ---------------------------------------------------------------------------
__global__ __launch_bounds__(256) void zb_init(u64* __restrict__ zb, int n) {
  int i = blockIdx.x * blockDim.x + threadIdx.x;
  if (i < n) zb[i] = ~0ull;
}

// ---------------------------------------------------------------------------
// Kernel 2: rasterize fragments -> 64-bit packed atomicMin z-buffer (L2 resident)
//   key = (depth_bits << 32) | fragIdx  reproduces reference segment_min
//   tie-breaking (min depth, then min fragment index) exactly & deterministically.
// ---------------------------------------------------------------------------
__global__ __launch_bounds__(256) void raster(const float* __restrict__ verts,
                                              const int* __restrict__ faces,
                                              u64* __restrict__ zb) {
  const int nimg = blockIdx.x / FBLK;
  const int f0 = (blockIdx.x % FBLK) * FB;
  const int tid = threadIdx.x;
  const float* __restrict__ vn = verts + (size_t)nimg * (Vv * 3);

  __shared__ int sIdx[FB * 3];
  __shared__ float4 sP[FB * 3]; // (px, py, z, 0) per corner

  // Prefetch next block's face indices (lowers to global_prefetch_b8 on gfx1250)
  if (tid == 0 && (f0 + FB) < Ff) {
    __builtin_prefetch(faces + (size_t)(f0 + FB) * 3, 0, 0);
  }

  // Stage the 24 contiguous face indices for this block into LDS.
  if (tid < FB * 3) {
#if defined(HAVE_ASYNC_LDS)
    __builtin_amdgcn_global_load_async_to_lds_b32(
        (__attribute__((address_space(1))) int*)(faces + (size_t)f0 * 3 + tid),
        (__attribute__((address_space(3))) int*)&sIdx[tid],
        /*offset=*/0, /*cpol=*/0);
#else
    sIdx[tid] = faces[(size_t)f0 * 3 + tid];
#endif
  }
#if defined(HAVE_ASYNC_LDS) && defined(HAVE_WAIT_ASYNC)
  __builtin_amdgcn_s_wait_asynccnt(0); // issuing wave drains its ASYNCcnt
#endif
  __syncthreads();

  // Project the block's 24 corners once.
  if (tid < FB * 3) {
    int vid = sIdx[tid];
    float x = vn[vid * 3 + 0];
    float y = vn[vid * 3 + 1];
    float z = vn[vid * 3 + 2];
    sP[tid] = make_float4(x / z, y / z, z, 0.f);
  }
  __syncthreads();

  // One thread per fragment; 8 faces * 144 fragments = 1152 per block.
  for (int i = tid; i < FB * Kk * Kk; i += 256) {
    int fi = i / (Kk * Kk);
    int r = i - fi * (Kk * Kk);
    int oy = r / Kk;
    int ox = r - oy * Kk;

    float4 P0 = sP[fi * 3 + 0];
    float4 P1 = sP[fi * 3 + 1];
    float4 P2 = sP[fi * 3 + 2];

    float bminx = floorf(fminf(P0.x, fminf(P1.x, P2.x)));
    float bminy = floorf(fminf(P0.y, fminf(P1.y, P2.y)));
    int bcx = (int)fminf(fmaxf(bminx, 0.f), (float)(Ww - Kk));
    int bcy = (int)fminf(fmaxf(bminy, 0.f), (float)(Hh - Kk));
    int px = bcx + ox;
    int py = bcy + oy;
    float pxf = (float)px;
    float pyf = (float)py;

    float area = (P1.x - P0.x) * (P2.y - P0.y) - (P1.y - P0.y) * (P2.x - P0.x);
    float w0 = ((P2.x - P1.x) * (pyf - P1.y) - (P2.y - P1.y) * (pxf - P1.x)) / area;
    float w1 = ((P0.x - P2.x) * (pyf - P2.y) - (P0.y - P2.y) * (pxf - P2.x)) / area;
    float w2 = 1.0f - w0 - w1;

    bool valid = (w0 >= 0.f) && (w1 >= 0.f) && (w2 >= 0.f) &&
                 (fabsf(area) > 1e-9f) &&
                 (P0.z > 0.f) && (P1.z > 0.f) && (P2.z > 0.f);
    if (valid) {
      float depth = w0 * P0.z + w1 * P1.z + w2 * P2.z; // > 0 when valid
      u32 frag = (u32)((f0 + fi) * (Kk * Kk) + r);     // < 2^21
      u64 key = ((u64)__float_as_uint(depth) << 32) | (u64)frag;
      atomicMin(&zb[(size_t)nimg * HW + (size_t)(py * Ww + px)], key);
    }
  }
}

// ---------------------------------------------------------------------------
// Kernel 3: resolve winners -> interpolate UV, apply mask transform, write out.
//   out layout: uvs [N,2,H,W] then mask [N,1,H,W]
// ---------------------------------------------------------------------------
__global__ __launch_bounds__(256) void resolve(const float* __restrict__ verts,
                                               const int* __restrict__ faces,
                                               const float* __restrict__ vals,
                                               const u64* __restrict__ zb,
                                               float* __restrict__ out) {
  int idx = blockIdx.x * blockDim.x + threadIdx.x;
  if (idx >= Nn * HW) return;
  int nimg = idx >> 18;        // HW == 2^18
  int pix = idx & (HW - 1);
  int y = pix >> 9;            // W == 512
  int x = pix & (Ww - 1);

  u64 key = zb[idx];
  float u = 0.f, v = 0.f;
  if (key != ~0ull) {
    u32 frag = (u32)key;
    int f = (int)(frag / (u32)(Kk * Kk));
    const float* vn = verts + (size_t)nimg * (Vv * 3);
    int i0 = faces[f * 3 + 0];
    int i1 = faces[f * 3 + 1];
    int i2 = faces[f * 3 + 2];
    float z0 = vn[i0 * 3 + 2], z1 = vn[i1 * 3 + 2], z2 = vn[i2 * 3 + 2];
    float x0 = vn[i0 * 3 + 0] / z0, y0 = vn[i0 * 3 + 1] / z0;
    float x1 = vn[i1 * 3 + 0] / z1, y1 = vn[i1 * 3 + 1] / z1;
    float x2 = vn[i2 * 3 + 0] / z2, y2 = vn[i2 * 3 + 1] / z2;
    float pxf = (float)x;
    float pyf = (float)y;
    float area = (x1 - x0) * (y2 - y0) - (y1 - y0) * (x2 - x0);
    float w0 = ((x2 - x1) * (pyf - y1) - (y2 - y1) * (pxf - x1)) / area;
    float w1 = ((x0 - x2) * (pyf - y2) - (y0 - y2) * (pxf - x2)) / area;
    float w2 = 1.0f - w0 - w1;
    u = w0 * vals[i0 * 2 + 0] + w1 * vals[i1 * 2 + 0] + w2 * vals[i2 * 2 + 0];
    v = w0 * vals[i0 * 2 + 1] + w1 * vals[i1 * 2 + 1] + w2 * vals[i2 * 2 + 1];
  }
  float m = ((u > 0.f) || (v > 0.f)) ? 1.f : 0.f;
  float ou = (u * 2.f - 1.f) * m - 10.f * (1.f - m);
  float ov = (v * 2.f - 1.f) * m - 10.f * (1.f - m);

  out[(size_t)(nimg * 2 + 0) * HW + pix] = ou;
  out[(size_t)(nimg * 2 + 1) * HW + pix] = ov;
  out[(size_t)Nn * 2 * HW + (size_t)nimg * HW + pix] = m;
}

extern "C" void kernel_launch(void* const* d_in, const int* in_sizes, int n_in,
                              void* d_out, int out_size, void* d_ws, size_t ws_size,
                              hipStream_t stream) {
  (void)in_sizes; (void)n_in; (void)out_size; (void)ws_size;
  const float* verts = (const float*)d_in[0];        // [N, V, 3] f32
  const int* faces = (const int*)d_in[1];            // [F, 3] i32
  const float* vals = (const float*)d_in[2];         // [V, 2] f32
  float* out = (float*)d_out;                        // uvs ++ mask
  u64* zb = (u64*)d_ws;                              // [N*H*W] u64 (16 MB)

  const int zn = Nn * HW;
  zb_init<<<(zn + 255) / 256, 256, 0, stream>>>(zb, zn);
  raster<<<Nn * FBLK, 256, 0, stream>>>(verts, faces, zb);
  resolve<<<(Nn * HW + 255) / 256, 256, 0, stream>>>(verts, faces, vals, zb, out);
}